// GINConvolution_36627481100824
// MI455X (gfx1250) — compile-verified
//
#include <hip/hip_runtime.h>

// GIN 2-layer forward for MI455X (gfx1250, wave32).
// Scatter-bound problem (~1.2GB traffic); dense 96x96 GEMM done with
// V_WMMA_F32_16X16X4_F32 (fp32, matches reference precision).

#define N_NODES   50000
#define N_EDGES   800000
#define D_FEAT    96
#define HIDDEN    96
#define N_CLASSES 2

typedef __attribute__((ext_vector_type(2))) float v2f;
typedef __attribute__((ext_vector_type(8))) float v8f;

// ---------------------------------------------------------------------------
// Zero the aggregation buffer (grid-stride).
__global__ void gin_zero_kernel(float* __restrict__ p, int n) {
  int i = blockIdx.x * blockDim.x + threadIdx.x;
  int stride = gridDim.x * blockDim.x;
  for (; i < n; i += stride) p[i] = 0.0f;
}

// ---------------------------------------------------------------------------
// Edge scatter-add: one wave32 per edge; lane L handles features L, L+32, L+64.
// Gather reads and atomic adds are 128B-coalesced per wave. aggr (19.2MB)
// fits in the 192MB L2, so the f32 atomics resolve at L2.
__global__ void gin_scatter_kernel(const float* __restrict__ feat,
                                   const int* __restrict__ src,
                                   const int* __restrict__ dst,
                                   float* __restrict__ aggr) {
  long long t = (long long)blockIdx.x * blockDim.x + threadIdx.x;
  int lane = (int)(t & 31);
  long long e = t >> 5;
  if (e >= N_EDGES) return;
  int s = src[e];
  int d = dst[e];
  const float* fs = feat + (long long)s * D_FEAT;
  float* ad = aggr + (long long)d * D_FEAT;
#pragma unroll
  for (int j = 0; j < 3; ++j) {
    int f = lane + j * 32;                 // 96 = 3 * 32 features
    atomicAdd(&ad[f], fs[f]);              // global_atomic_add_f32
  }
}

// ---------------------------------------------------------------------------
// Layer-1 GEMM with WMMA: h = relu((x + aggr) @ W1 + b1).
// One wave computes one 16x16 tile of h; K=96 swept in steps of 4 using
// V_WMMA_F32_16X16X4_F32 (fp32 A/B/C). 50000 rows = 3125 exact M-tiles,
// 96 cols = 6 N-tiles -> 18750 waves; guard is wave-uniform so EXEC is
// all-ones inside every WMMA.
__global__ void gin_gemm1_wmma_kernel(const float* __restrict__ x,
                                      const float* __restrict__ aggr,
                                      const float* __restrict__ W1,   // [K=96, N=96]
                                      const float* __restrict__ b1,   // [96]
                                      float* __restrict__ h) {
  const int tilesM = N_NODES / 16;   // 3125
  const int tilesN = HIDDEN / 16;    // 6
  int wave = blockIdx.x * (blockDim.x >> 5) + (threadIdx.x >> 5);
  if (wave >= tilesM * tilesN) return;   // wave-uniform exit
  int tm = wave / tilesN;
  int tn = wave % tilesN;

  int lane = threadIdx.x & 31;
  int half = lane >> 4;   // 0: lanes 0-15, 1: lanes 16-31
  int l    = lane & 15;

  int row = tm * 16 + l;       // A-fragment row for this lane
  int col = tn * 16 + l;       // B/C/D-fragment column for this lane

  const float* xr = x    + (long long)row * D_FEAT;
  const float* ar = aggr + (long long)row * D_FEAT;

  v8f c = {};   // 16x16 f32 accumulator, 8 VGPRs

#pragma unroll
  for (int k = 0; k < D_FEAT; k += 4) {
    // A 16x4 f32 fragment: VGPR0 = K {k+2*half}, VGPR1 = K {k+2*half+1}
    int ka = k + 2 * half;
    v2f a, b;
    a.x = xr[ka]     + ar[ka];        // fuse h_in = x + aggr
    a.y = xr[ka + 1] + ar[ka + 1];
    // B 4x16 f32 fragment: row K striped across lanes (same half-split as A)
    b.x = W1[(long long)ka       * HIDDEN + col];
    b.y = W1[(long long)(ka + 1) * HIDDEN + col];
    c = __builtin_amdgcn_wmma_f32_16x16x4_f32(
            /*neg_a=*/false, a, /*neg_b=*/false, b,
            /*c_mod=*/(short)0, c, /*reuse_a=*/false, /*reuse_b=*/false);
  }

  // Epilogue: bias + ReLU, store. C/D layout: VGPR v, lane half -> M = v+8*half,
  // N = l (same column for all 8 VGPRs).
  float bias = b1[col];
#pragma unroll
  for (int v = 0; v < 8; ++v) {
    int r = tm * 16 + v + 8 * half;
    float val = c[v] + bias;
    val = val > 0.0f ? val : 0.0f;
    h[(long long)r * HIDDEN + col] = val;
  }
}

// ---------------------------------------------------------------------------
// Layer-2 output: out = (h + aggr2) @ W2 + b2 with N_CLASSES = 2.
// N=2 wastes WMMA's 16-wide tiles; a VALU dot (2 FMAs x 96) per node is the
// right tool — still bound by reading h+aggr2 (38MB).
__global__ void gin_out_kernel(const float* __restrict__ h,
                               const float* __restrict__ aggr,
                               const float* __restrict__ W2,   // [96, 2]
                               const float* __restrict__ b2,   // [2]
                               float* __restrict__ out) {
  int n = blockIdx.x * blockDim.x + threadIdx.x;
  if (n >= N_NODES) return;
  const float* hr = h    + (long long)n * HIDDEN;
  const float* ar = aggr + (long long)n * HIDDEN;
  float acc0 = b2[0];
  float acc1 = b2[1];
#pragma unroll 8
  for (int k = 0; k < HIDDEN; ++k) {
    float v = hr[k] + ar[k];
    acc0 = fmaf(v, W2[k * 2 + 0], acc0);
    acc1 = fmaf(v, W2[k * 2 + 1], acc1);
  }
  out[(long long)n * 2 + 0] = acc0;
  out[(long long)n * 2 + 1] = acc1;
}

// ---------------------------------------------------------------------------
extern "C" void kernel_launch(void* const* d_in, const int* in_sizes, int n_in,
                              void* d_out, int out_size, void* d_ws, size_t ws_size,
                              hipStream_t stream) {
  const float* x  = (const float*)d_in[0];
  const int*   ei = (const int*)d_in[1];   // [2, N_EDGES] flat: src then dst
  const float* W1 = (const float*)d_in[2];
  const float* b1 = (const float*)d_in[3];
  const float* W2 = (const float*)d_in[4];
  const float* b2 = (const float*)d_in[5];
  float* out = (float*)d_out;

  const int* src = ei;
  const int* dst = ei + N_EDGES;

  float* aggr = (float*)d_ws;                          // 50000*96 floats (19.2MB)
  float* h    = aggr + (size_t)N_NODES * HIDDEN;       // 50000*96 floats (19.2MB)

  const int nfeat = N_NODES * HIDDEN;

  // Layer 1
  gin_zero_kernel<<<2048, 256, 0, stream>>>(aggr, nfeat);
  long long sthreads = (long long)N_EDGES * 32;        // one wave32 per edge
  gin_scatter_kernel<<<(unsigned)((sthreads + 255) / 256), 256, 0, stream>>>(
      x, src, dst, aggr);
  const int nTiles = (N_NODES / 16) * (HIDDEN / 16);   // 18750 waves
  gin_gemm1_wmma_kernel<<<(nTiles + 7) / 8, 256, 0, stream>>>(
      x, aggr, W1, b1, h);

  // Layer 2 (reuse aggr buffer)
  gin_zero_kernel<<<2048, 256, 0, stream>>>(aggr, nfeat);
  gin_scatter_kernel<<<(unsigned)((sthreads + 255) / 256), 256, 0, stream>>>(
      h, src, dst, aggr);
  gin_out_kernel<<<(N_NODES + 255) / 256, 256, 0, stream>>>(
      h, aggr, W2, b2, out);
}